// RelativeEventEmbedding_62242666054243
// MI455X (gfx1250) — compile-verified
//
#include <hip/hip_runtime.h>
#include <hip/hip_bf16.h>
#include <math.h>

typedef __attribute__((ext_vector_type(2))) float v2f;
typedef __attribute__((ext_vector_type(4))) float v4f;
typedef __attribute__((ext_vector_type(8))) float v8f;

#define BB 4
#define SS 2048
#define FF 256
#define DD 1024
#define NSC 5
#define LDP 68      // padded LDS row stride (floats): 68%64=4 -> conflict-free frag reads
#define PSTR 258    // padded stride for 256-wide staged tiles

// ---- workspace layout (float elements) ----
static constexpr size_t OG   = 0;                                  // gram  B*S*S
static constexpr size_t OQ   = OG   + (size_t)BB*SS*SS;            // q     B*S*F
static constexpr size_t OK_  = OQ   + (size_t)BB*SS*FF;            // k     B*S*F
static constexpr size_t OSQ  = OK_  + (size_t)BB*SS*FF;            // sum x^2 per row
static constexpr size_t ORN  = OSQ  + (size_t)BB*SS;               // 1/max(norm,1e-12)
static constexpr size_t ORM  = ORN  + (size_t)BB*SS;               // row max (ordered uint)
static constexpr size_t ORS  = ORM  + (size_t)BB*SS;               // row sum exp
static constexpr size_t OPE  = ORS  + (size_t)BB*SS;               // pe    B*S*F
static constexpr size_t OST  = OPE  + (size_t)BB*SS*FF;            // stats: sum,sumsq,minenc,maxenc
static constexpr size_t OPRM = OST  + 8;                           // lo,hi,mn,irng,a,inv2sig2

__device__ __forceinline__ v8f wmma_f32(v2f a, v2f b, v8f c) {
  // V_WMMA_F32_16X16X4_F32 : D = A(16x4) * B(4x16) + C(16x16), fp32
  return __builtin_amdgcn_wmma_f32_16x16x4_f32(false, a, false, b, (short)0, c, false, false);
}
__device__ __forceinline__ v8f vzero8() {
  v8f z = {0.f,0.f,0.f,0.f,0.f,0.f,0.f,0.f};
  return z;
}
__device__ __forceinline__ unsigned fenc(float f) {     // order-preserving float->uint
  unsigned u = __float_as_uint(f);
  return (u & 0x80000000u) ? ~u : (u | 0x80000000u);
}
__device__ __forceinline__ float fdec(unsigned u) {
  return __uint_as_float((u & 0x80000000u) ? (u & 0x7fffffffu) : ~u);
}

// stage a 64x64 fp32 tile (row stride ldsrc) into LDS with stride LDP; 256 threads
__device__ __forceinline__ void stage64(const float* __restrict__ src, int ldsrc,
                                        float* __restrict__ dst, int tid) {
  int r  = tid >> 2;
  int c0 = (tid & 3) * 16;
  const float* s = src + (size_t)r * ldsrc + c0;
  float* d = dst + r * LDP + c0;
#pragma unroll
  for (int j = 0; j < 16; j += 4) *(v4f*)(d + j) = *(const v4f*)(s + j);
}
// stage a 64x64 tile transposed into LDS (dst[n][k] = src[k][n])
__device__ __forceinline__ void stage64t(const float* __restrict__ src, int ldsrc,
                                         float* __restrict__ dst, int tid) {
  int r  = tid >> 2;
  int c0 = (tid & 3) * 16;
  const float* s = src + (size_t)r * ldsrc + c0;
#pragma unroll
  for (int j = 0; j < 16; ++j) dst[(c0 + j) * LDP + r] = s[j];
}

// 64x64 tile of Arows(64xFF) * Brows(64xFF)^T via f32 WMMA; wave w owns subtiles
// (sm=w&3, sn=(w>>2)*2 .. +1). acc0/acc1 are the two 16x16 accumulators.
__device__ __forceinline__ void mm64(const float* __restrict__ Arows,
                                     const float* __restrict__ Brows,
                                     float* lA, float* lB, int tid,
                                     v8f& acc0, v8f& acc1) {
  int w = tid >> 5, lane = tid & 31, lh = lane >> 4, lm = lane & 15;
  int sm = w & 3, sn = (w >> 2) * 2;
  for (int kc = 0; kc < 4; ++kc) {
    __syncthreads();
    stage64(Arows + kc * 64, FF, lA, tid);
    stage64(Brows + kc * 64, FF, lB, tid);
    __syncthreads();
#pragma unroll
    for (int s8 = 0; s8 < 16; ++s8) {
      int kk = s8 * 4 + lh * 2;
      v2f a  = *(const v2f*)(lA + (sm * 16 + lm) * LDP + kk);
      v2f b0 = *(const v2f*)(lB + (sn * 16 + lm) * LDP + kk);
      v2f b1 = *(const v2f*)(lB + ((sn + 1) * 16 + lm) * LDP + kk);
      acc0 = wmma_f32(a, b0, acc0);
      acc1 = wmma_f32(a, b1, acc1);
    }
  }
}

// ---------------- kernel 0: init reduction buffers ----------------
__global__ __launch_bounds__(256) void k_init(float* ws) {
  int i = blockIdx.x * 256 + threadIdx.x;
  if (i < BB * SS) {
    ((unsigned*)(ws + ORM))[i] = 0u;   // enc of -inf-ish
    ws[ORS + i] = 0.f;
  }
  if (i == 0) {
    ws[OST + 0] = 0.f; ws[OST + 1] = 0.f;
    ((unsigned*)(ws + OST))[2] = 0xFFFFFFFFu;  // min init
    ((unsigned*)(ws + OST))[3] = 0u;           // max init
  }
}

// ---------------- kernel 1: per-row sum(x^2) and 1/norm ----------------
__global__ __launch_bounds__(256) void k_sqnorm(const float* __restrict__ x, float* ws) {
  int w = threadIdx.x >> 5, lane = threadIdx.x & 31;
  int row = blockIdx.x * 8 + w;                   // 0 .. B*S-1
  const float* xr = x + (size_t)row * FF;
  float s = 0.f;
#pragma unroll
  for (int i = 0; i < FF; i += 32) { float v = xr[i + lane]; s += v * v; }
  for (int m = 16; m; m >>= 1) s += __shfl_xor(s, m, 32);
  if (lane == 0) {
    ws[OSQ + row] = s;
    float n = sqrtf(s);
    ws[ORN + row] = 1.f / fmaxf(n, 1e-12f);
  }
}

// ---------------- kernel 2: q = xWq+bq, k = xWk+bk (z selects) ----------------
__global__ __launch_bounds__(256) void k_qk(const float* __restrict__ x,
                                            const float* __restrict__ Wq, const float* __restrict__ bq,
                                            const float* __restrict__ Wk, const float* __restrict__ bk,
                                            float* ws) {
  __shared__ __align__(16) float xs[64 * LDP];
  __shared__ __align__(16) float wt[64 * LDP];   // transposed W slice: wt[n][k]
  int tid = threadIdx.x, w = tid >> 5, lane = tid & 31, lh = lane >> 4, lm = lane & 15;
  int row0 = blockIdx.x * 64;                    // flat row
  int n0   = blockIdx.y * 64;                    // output col
  const float* W    = blockIdx.z ? Wk : Wq;
  const float* bias = blockIdx.z ? bk : bq;
  float* out = ws + (blockIdx.z ? OK_ : OQ);
  int sm = w & 3, sn = (w >> 2) * 2;
  v8f acc0 = vzero8(), acc1 = vzero8();
  for (int kc = 0; kc < 4; ++kc) {
    __syncthreads();
    stage64(x + (size_t)row0 * FF + kc * 64, FF, xs, tid);
    stage64t(W + (size_t)(kc * 64) * FF + n0, FF, wt, tid);
    __syncthreads();
#pragma unroll
    for (int s8 = 0; s8 < 16; ++s8) {
      int kk = s8 * 4 + lh * 2;
      v2f a  = *(const v2f*)(xs + (sm * 16 + lm) * LDP + kk);
      v2f b0 = *(const v2f*)(wt + (sn * 16 + lm) * LDP + kk);
      v2f b1 = *(const v2f*)(wt + ((sn + 1) * 16 + lm) * LDP + kk);
      acc0 = wmma_f32(a, b0, acc0);
      acc1 = wmma_f32(a, b1, acc1);
    }
  }
#pragma unroll
  for (int r = 0; r < 8; ++r) {
    int gr = row0 + sm * 16 + r + lh * 8;
    int c0 = n0 + sn * 16 + lm, c1 = c0 + 16;
    out[(size_t)gr * FF + c0] = acc0[r] + bias[c0];
    out[(size_t)gr * FF + c1] = acc1[r] + bias[c1];
  }
}

// ---------------- kernel 3: gram G = x x^T (symmetric) + eucl stats ----------------
__global__ __launch_bounds__(256) void k_gram(const float* __restrict__ x, float* ws) {
  int bx = blockIdx.x, by = blockIdx.y, b = blockIdx.z;
  if (by < bx) return;                           // symmetry: compute upper triangle only
  __shared__ __align__(16) float lA[64 * LDP];
  __shared__ __align__(16) float lB[64 * LDP];
  __shared__ float red[32];
  int tid = threadIdx.x, w = tid >> 5, lane = tid & 31, lh = lane >> 4, lm = lane & 15;
  int sm = w & 3, sn = (w >> 2) * 2;
  int s0 = bx * 64, t0 = by * 64;
  const float* xb = x + (size_t)b * SS * FF;
  v8f acc0 = vzero8(), acc1 = vzero8();
  mm64(xb + (size_t)s0 * FF, xb + (size_t)t0 * FF, lA, lB, tid, acc0, acc1);

  float* Gp = ws + OG + (size_t)b * SS * SS;
  const float* sqs = ws + OSQ + (size_t)b * SS;
  float mult = (bx == by) ? 1.f : 2.f;
  float lsum = 0.f, lsq = 0.f, lmn = 3.4e38f, lmx = -3.4e38f;
#pragma unroll
  for (int half = 0; half < 2; ++half) {
    int ct = t0 + (sn + half) * 16 + lm;
#pragma unroll
    for (int r = 0; r < 8; ++r) {
      float g = half ? acc1[r] : acc0[r];
      int rs = s0 + sm * 16 + r + lh * 8;
      Gp[(size_t)rs * SS + ct] = g;
      if (bx != by) Gp[(size_t)ct * SS + rs] = g;
      float d2 = sqs[rs] + sqs[ct] - 2.f * g;
      float e = sqrtf(fmaxf(d2, 0.f));
      lsum += mult * e; lsq += mult * e * e;
      lmn = fminf(lmn, e); lmx = fmaxf(lmx, e);
    }
  }
  for (int m = 16; m; m >>= 1) {
    lsum += __shfl_xor(lsum, m, 32);
    lsq  += __shfl_xor(lsq,  m, 32);
    lmn = fminf(lmn, __shfl_xor(lmn, m, 32));
    lmx = fmaxf(lmx, __shfl_xor(lmx, m, 32));
  }
  if (lane == 0) { red[w*4+0]=lsum; red[w*4+1]=lsq; red[w*4+2]=lmn; red[w*4+3]=lmx; }
  __syncthreads();
  if (tid == 0) {
    float a=0.f,q=0.f,mn=3.4e38f,mx=-3.4e38f;
    for (int i=0;i<8;++i){ a+=red[i*4]; q+=red[i*4+1]; mn=fminf(mn,red[i*4+2]); mx=fmaxf(mx,red[i*4+3]); }
    atomicAdd(ws + OST + 0, a);
    atomicAdd(ws + OST + 1, q);
    atomicMin((unsigned*)(ws + OST) + 2, fenc(mn));
    atomicMax((unsigned*)(ws + OST) + 3, fenc(mx));
  }
}

// ---------------- kernel 4: row max of scores ----------------
__global__ __launch_bounds__(256) void k_rowmax(float* ws) {
  __shared__ __align__(16) float lA[64 * LDP];
  __shared__ __align__(16) float lB[64 * LDP];
  int tid = threadIdx.x, w = tid >> 5, lane = tid & 31, lh = lane >> 4, lm = lane & 15;
  int sm = w & 3;
  int s0 = blockIdx.x * 64, t0 = blockIdx.y * 64, b = blockIdx.z;
  v8f acc0 = vzero8(), acc1 = vzero8();
  mm64(ws + OQ + (size_t)(b * SS + s0) * FF, ws + OK_ + (size_t)(b * SS + t0) * FF,
       lA, lB, tid, acc0, acc1);
  unsigned* rmax = (unsigned*)(ws + ORM) + b * SS + s0;
#pragma unroll
  for (int r = 0; r < 8; ++r) {
    float m = fmaxf(acc0[r], acc1[r]) * 0.0625f;     // /sqrt(F)=1/16
    for (int mm = 8; mm; mm >>= 1) m = fmaxf(m, __shfl_xor(m, mm, 16));
    if (lm == 0) atomicMax(&rmax[sm * 16 + r + lh * 8], fenc(m));
  }
}

// ---------------- kernel 5: row sum of exp(score - rowmax) ----------------
__global__ __launch_bounds__(256) void k_rowsum(float* ws) {
  __shared__ __align__(16) float lA[64 * LDP];
  __shared__ __align__(16) float lB[64 * LDP];
  __shared__ float mrs[64];
  int tid = threadIdx.x, w = tid >> 5, lane = tid & 31, lh = lane >> 4, lm = lane & 15;
  int sm = w & 3;
  int s0 = blockIdx.x * 64, t0 = blockIdx.y * 64, b = blockIdx.z;
  if (tid < 64) mrs[tid] = fdec(((const unsigned*)(ws + ORM))[b * SS + s0 + tid]);
  v8f acc0 = vzero8(), acc1 = vzero8();
  mm64(ws + OQ + (size_t)(b * SS + s0) * FF, ws + OK_ + (size_t)(b * SS + t0) * FF,
       lA, lB, tid, acc0, acc1);
  float* rsum = ws + ORS + b * SS + s0;
#pragma unroll
  for (int r = 0; r < 8; ++r) {
    int row16 = sm * 16 + r + lh * 8;
    float m = mrs[row16];
    float e = __expf(acc0[r] * 0.0625f - m) + __expf(acc1[r] * 0.0625f - m);
    for (int mm = 8; mm; mm >>= 1) e += __shfl_xor(e, mm, 16);
    if (lm == 0) atomicAdd(&rsum[row16], e);
  }
}

// ---------------- kernel 6: finalize global stats -> params ----------------
__global__ void k_finalize(const float* __restrict__ alpha, const float* __restrict__ sigma_w,
                           float* ws) {
  if (threadIdx.x == 0 && blockIdx.x == 0) {
    const float N = (float)((size_t)BB * SS * SS);
    float sum = ws[OST], sumsq = ws[OST + 1];
    float mean = sum / N;
    float var  = (sumsq - sum * (sum / N)) / (N - 1.f);    // ddof=1
    float sd   = sqrtf(fmaxf(var, 0.f));
    float lo = mean - 2.f * sd, hi = mean + 2.f * sd;
    float emin = fdec(((unsigned*)(ws + OST))[2]);
    float emax = fdec(((unsigned*)(ws + OST))[3]);
    float mn = fminf(fmaxf(emin, lo), hi);                  // clip is monotone clamp
    float mx = fminf(fmaxf(emax, lo), hi);
    ws[OPRM + 0] = lo;
    ws[OPRM + 1] = hi;
    ws[OPRM + 2] = mn;
    ws[OPRM + 3] = 1.f / (mx - mn + 1e-6f);
    ws[OPRM + 4] = 1.f / (1.f + expf(-alpha[0]));           // sigmoid
    float sig = log1pf(expf(sigma_w[0])) + 0.001f;          // softplus + eps
    ws[OPRM + 5] = 1.f / (2.f * sig * sig);
  }
}

// ---------------- kernel 7: flash-style pe = E @ x ----------------
// block: 32 s-rows; loops over all t in chunks of 64; recomputes scores via WMMA,
// builds E = exp(score-m-hyb/2sig^2)/l in LDS, accumulates pe via WMMA.
__global__ __launch_bounds__(256) void k_pe(const float* __restrict__ x, float* ws) {
  __shared__ __align__(16) float qs[32 * PSTR];
  __shared__ __align__(16) float kx[64 * LDP];
  __shared__ __align__(16) float Es[32 * LDP];
  __shared__ float sqt[64], rnt[64];
  __shared__ float mrow[32], lirow[32], sqs_s[32], rns_s[32], prm[8];
  int tid = threadIdx.x, w = tid >> 5, lane = tid & 31, lh = lane >> 4, lm = lane & 15;
  int b = blockIdx.y, s0 = blockIdx.x * 32;
  int rg = w & 1, cgp = w >> 1;                   // wave role: row group / col group
  const float* xb = x + (size_t)b * SS * FF;
  const float* qrow = ws + OQ + (size_t)(b * SS + s0) * FF;
  for (int i = tid; i < 32 * FF; i += 256) {
    int r = i >> 8, c = i & 255;
    qs[r * PSTR + c] = qrow[(size_t)r * FF + c];
  }
  if (tid < 32) {
    int row = b * SS + s0 + tid;
    mrow[tid]  = fdec(((const unsigned*)(ws + ORM))[row]);
    lirow[tid] = 1.f / ws[ORS + row];
    sqs_s[tid] = ws[OSQ + row];
    rns_s[tid] = ws[ORN + row];
  }
  if (tid < 6) prm[tid] = ws[OPRM + tid];
  v8f acc[4]; for (int f = 0; f < 4; ++f) acc[f] = vzero8();
  const float* Gb = ws + OG + (size_t)b * SS * SS;

  for (int tc = 0; tc < SS / 64; ++tc) {
    int t0 = tc * 64;
    __syncthreads();
    if (tid < 64) { sqt[tid] = ws[OSQ + b * SS + t0 + tid]; rnt[tid] = ws[ORN + b * SS + t0 + tid]; }
    // scores: wave computes 16x16 subtile (rows rg*16.., cols t0+cgp*16..)
    v8f sc = vzero8();
    for (int kc = 0; kc < 4; ++kc) {
      __syncthreads();
      stage64(ws + OK_ + (size_t)(b * SS + t0) * FF + kc * 64, FF, kx, tid);
      __syncthreads();
#pragma unroll
      for (int s8 = 0; s8 < 16; ++s8) {
        int kk = s8 * 4 + lh * 2;
        v2f a  = *(const v2f*)(qs + (rg * 16 + lm) * PSTR + kc * 64 + kk);
        v2f bb = *(const v2f*)(kx + (cgp * 16 + lm) * LDP + kk);
        sc = wmma_f32(a, bb, sc);
      }
    }
    // convert scores -> E tile in LDS
    float lo = prm[0], hi = prm[1], mn = prm[2], irng = prm[3], aa = prm[4], i2s = prm[5];
#pragma unroll
    for (int r = 0; r < 8; ++r) {
      int lr = rg * 16 + r + lh * 8;              // local s row (0..31)
      int tl = cgp * 16 + lm;                     // local t (0..63)
      float g = Gb[(size_t)(s0 + lr) * SS + (t0 + tl)];
      float cosv = (g * rns_s[lr] * rnt[tl] + 1.f) * 0.5f;
      float d2 = sqs_s[lr] + sqt[tl] - 2.f * g;
      float eu = sqrtf(fmaxf(d2, 0.f));
      float en = (fminf(fmaxf(eu, lo), hi) - mn) * irng;
      float hyb = aa * cosv + (1.f - aa) * (1.f - en);
      float E = __expf(sc[r] * 0.0625f - mrow[lr] - hyb * i2s) * lirow[lr];
      Es[lr * LDP + tl] = E;
    }
    // pe accumulation over this t-chunk, feature slices of 64
    for (int fs = 0; fs < 4; ++fs) {
      __syncthreads();
      stage64(xb + (size_t)t0 * FF + fs * 64, FF, kx, tid);
      __syncthreads();
#pragma unroll
      for (int s8 = 0; s8 < 16; ++s8) {
        int kk = s8 * 4 + lh * 2;
        v2f a = *(const v2f*)(Es + (rg * 16 + lm) * LDP + kk);
        v2f bb;
        bb.x = kx[kk * LDP + cgp * 16 + lm];
        bb.y = kx[(kk + 1) * LDP + cgp * 16 + lm];
        acc[fs] = wmma_f32(a, bb, acc[fs]);
      }
    }
  }
  float* pe = ws + OPE + (size_t)(b * SS + s0) * FF;
#pragma unroll
  for (int fs = 0; fs < 4; ++fs)
#pragma unroll
    for (int r = 0; r < 8; ++r)
      pe[(size_t)(rg * 16 + r + lh * 8) * FF + fs * 64 + cgp * 16 + lm] = acc[fs][r];
}

// ---------------- kernel 8: out = LN(pe Wproj + bproj) + LN(rel_enc), fused ----------------
// block = 16 rows x full D=1024; wave w owns cols w*128..+127 (8 subtiles).
__global__ __launch_bounds__(256) void k_out(const float* __restrict__ x,
                                             const float* __restrict__ Wrpe, const float* __restrict__ brpe,
                                             const float* __restrict__ g1, const float* __restrict__ b1,
                                             const float* __restrict__ Wproj, const float* __restrict__ bproj,
                                             const float* __restrict__ g2, const float* __restrict__ b2,
                                             const float* __restrict__ sw,
                                             const float* __restrict__ ws, float* __restrict__ out) {
  __shared__ __align__(16) float pes[16 * PSTR];
  __shared__ float cc[16][3];
  __shared__ float redbuf[8][16][2];
  __shared__ float relred[16][16][2];
  __shared__ float rstat[16][4];
  int tid = threadIdx.x, w = tid >> 5, lane = tid & 31, lh = lane >> 4, lm = lane & 15;
  int gr0 = blockIdx.x * 16;
  const float* pe = ws + OPE + (size_t)gr0 * FF;
  for (int i = tid; i < 16 * FF; i += 256) {
    int r = i >> 8, c = i & 255;
    pes[r * PSTR + c] = pe[(size_t)r * FF + c];
  }
  // rel-pos scalars c0,c1,c2 per row
  if (tid < 16) {
    int grow = gr0 + tid;
    int s = grow & (SS - 1);
    const float* xr = x + (size_t)grow * FF;
    float c0 = 0.f, c1 = 0.f, c2 = 0.f, inv = 0.5f;
#pragma unroll
    for (int sc = 1; sc <= NSC; ++sc) {
      if (s >= sc) {
        const float* xp = x + (size_t)(grow - sc) * FF;
        float dd = xp[0] - xr[0];
        float de = xp[FF - 1] - xr[FF - 1];
        c0 += dd; c1 += de;
        c2 += sw[sc - 1] * __expf(-fabsf(dd) * inv);
      }
      inv *= 0.5f;
    }
    cc[tid][0] = c0; cc[tid][1] = c1; cc[tid][2] = c2;
  }
  __syncthreads();

  v8f acc[8]; for (int cg = 0; cg < 8; ++cg) acc[cg] = vzero8();
  for (int s8 = 0; s8 < 64; ++s8) {
    int kk = s8 * 4 + lh * 2;
    v2f a = *(const v2f*)(pes + lm * PSTR + kk);
    const float* wp0 = Wproj + (size_t)kk * DD + w * 128 + lm;
    const float* wp1 = wp0 + DD;
#pragma unroll
    for (int cg = 0; cg < 8; ++cg) {
      v2f bb; bb.x = wp0[cg * 16]; bb.y = wp1[cg * 16];
      acc[cg] = wmma_f32(a, bb, acc[cg]);
    }
  }
  // layernorm stats for pe-projection rows
#pragma unroll
  for (int r = 0; r < 8; ++r) {
    int row16 = r + lh * 8;
    float ps = 0.f, pq = 0.f;
#pragma unroll
    for (int cg = 0; cg < 8; ++cg) {
      float v = acc[cg][r] + bproj[w * 128 + cg * 16 + lm];
      ps += v; pq += v * v;
    }
    for (int m = 8; m; m >>= 1) { ps += __shfl_xor(ps, m, 16); pq += __shfl_xor(pq, m, 16); }
    if (lm == 0) { redbuf[w][row16][0] = ps; redbuf[w][row16][1] = pq; }
  }
  // layernorm stats for rel_enc rows (recompute cheap)
  {
    int row = tid >> 4, seg = tid & 15;
    float rs_ = 0.f, rq_ = 0.f;
    float c0 = cc[row][0], c1 = cc[row][1], c2 = cc[row][2];
    for (int d = seg * 64; d < seg * 64 + 64; ++d) {
      float rv = c0 * Wrpe[d] + c1 * Wrpe[DD + d] + c2 * Wrpe[2 * DD + d] + 5.f * brpe[d];
      rs_ += rv; rq_ += rv * rv;
    }
    relred[row][seg][0] = rs_; relred[row][seg][1] = rq_;
  }
  __syncthreads();
  if (tid < 16) {
    float s = 0.f, q = 0.f;
    for (int ww = 0; ww < 8; ++ww) { s += redbuf[ww][tid][0]; q += redbuf[ww][tid][1]; }
    float mu = s / (float)DD;
    rstat[tid][0] = mu;
    rstat[tid][1] = rsqrtf(q / (float)DD - mu * mu + 1e-5f);
    float s2 = 0.f, q2 = 0.f;
    for (int g = 0; g < 16; ++g) { s2 += relred[tid][g][0]; q2 += relred[tid][g][1]; }
    float mur = s2 / (float)DD;
    rstat[tid][2] = mur;
    rstat[tid][3] = rsqrtf(q2 / (float)DD - mur * mur + 1e-5f);
  }
  __syncthreads();
#pragma unroll
  for (int cg = 0; cg < 8; ++cg) {
    int n = w * 128 + cg * 16 + lm;
    float w0 = Wrpe[n], w1 = Wrpe[DD + n], w2 = Wrpe[2 * DD + n], bb5 = 5.f * brpe[n];
    float g1n = g1[n], b1n = b1[n], g2n = g2[n], b2n = b2[n], bpn = bproj[n];
#pragma unroll
    for (int r = 0; r < 8; ++r) {
      int row16 = r + lh * 8;
      float v = acc[cg][r] + bpn;
      float lnpe = (v - rstat[row16][0]) * rstat[row16][1] * g2n + b2n;
      float rel = cc[row16][0] * w0 + cc[row16][1] * w1 + cc[row16][2] * w2 + bb5;
      float lnrel = (rel - rstat[row16][2]) * rstat[row16][3] * g1n + b1n;
      out[(size_t)(gr0 + row16) * DD + n] = lnpe + lnrel;
    }
  }
}

extern "C" void kernel_launch(void* const* d_in, const int* in_sizes, int n_in,
                              void* d_out, int out_size, void* d_ws, size_t ws_size,
                              hipStream_t stream) {
  const float* x       = (const float*)d_in[0];
  const float* Wrpe    = (const float*)d_in[1];
  const float* brpe    = (const float*)d_in[2];
  const float* ln1g    = (const float*)d_in[3];
  const float* ln1b    = (const float*)d_in[4];
  const float* Wproj   = (const float*)d_in[5];
  const float* bproj   = (const float*)d_in[6];
  const float* ln2g    = (const float*)d_in[7];
  const float* ln2b    = (const float*)d_in[8];
  const float* alpha   = (const float*)d_in[9];
  const float* sigma_w = (const float*)d_in[10];
  const float* Wq      = (const float*)d_in[11];
  const float* bq      = (const float*)d_in[12];
  const float* Wk      = (const float*)d_in[13];
  const float* bk      = (const float*)d_in[14];
  const float* sw      = (const float*)d_in[15];
  float* ws  = (float*)d_ws;
  float* out = (float*)d_out;

  k_init   <<<dim3((BB*SS + 255) / 256), 256, 0, stream>>>(ws);
  k_sqnorm <<<dim3(BB * SS / 8), 256, 0, stream>>>(x, ws);
  k_qk     <<<dim3(BB * SS / 64, FF / 64, 2), 256, 0, stream>>>(x, Wq, bq, Wk, bk, ws);
  k_gram   <<<dim3(SS / 64, SS / 64, BB), 256, 0, stream>>>(x, ws);
  k_rowmax <<<dim3(SS / 64, SS / 64, BB), 256, 0, stream>>>(ws);
  k_rowsum <<<dim3(SS / 64, SS / 64, BB), 256, 0, stream>>>(ws);
  k_finalize<<<1, 32, 0, stream>>>(alpha, sigma_w, ws);
  k_pe     <<<dim3(SS / 32, BB), 256, 0, stream>>>(x, ws);
  k_out    <<<dim3(BB * SS / 16), 256, 0, stream>>>(x, Wrpe, brpe, ln1g, ln1b,
                                                    Wproj, bproj, ln2g, ln2b, sw, ws, out);
}